// ScaleDotProductAttention_75419625717958
// MI455X (gfx1250) — compile-verified
//
#include <hip/hip_runtime.h>
#include <hip/hip_bf16.h>

typedef __attribute__((ext_vector_type(2))) float v2f;
typedef __attribute__((ext_vector_type(8))) float v8f;

// Problem constants from the reference setup.
#define BB 4
#define HH 8
#define LL 768
#define DD 64
#define NUM 100

// ---------------------------------------------------------------------------
// Kernel 1: score = (q @ k.view(B,H,D,L)) * 1/sqrt(D), fp32 WMMA 16x16x4.
// One wave computes one 16x16 tile of score for one (b,h). 4 waves / block.
// A-matrix layout (16x4 f32): lane = (halfK<<4)|m ; holds K = 2*halfK+{0,1}.
// B-matrix layout (4x16 f32): lane = (halfK<<4)|n ; holds K = 2*halfK+{0,1}.
// C/D layout  (16x16 f32)   : VGPR r -> M = r + 8*(lane>>4), N = lane&15.
// ---------------------------------------------------------------------------
__global__ void __launch_bounds__(128)
qk_gemm_wmma_f32(const float* __restrict__ q, const float* __restrict__ k,
                 float* __restrict__ score)
{
    constexpr int TILES = LL / 16;                 // 48 tiles per dim
    const int lane    = threadIdx.x & 31;
    const int waveId  = blockIdx.x * 4 + (threadIdx.x >> 5);
    const int bh      = blockIdx.y;                // 0..B*H-1
    const int tileM   = (waveId / TILES) * 16;
    const int tileN   = (waveId % TILES) * 16;

    const float* qbh = q + (size_t)bh * LL * DD;
    const float* kbh = k + (size_t)bh * LL * DD;   // reinterpret as [DD][LL]
    float*       sbh = score + (size_t)bh * LL * LL;

    const int m16   = lane & 15;                   // M (for A) / N (for B)
    const int kpair = (lane >> 4) << 1;            // 0 or 2

    v8f acc = {};
#pragma unroll
    for (int k0 = 0; k0 < DD; k0 += 4) {
        // A: q[tileM+m16][k0+kpair], q[tileM+m16][k0+kpair+1]  (contiguous)
        v2f a = *(const v2f*)(qbh + (size_t)(tileM + m16) * DD + (k0 + kpair));
        // B: k_t[k0+kpair][tileN+m16], k_t[k0+kpair+1][tileN+m16] (L-strided)
        const float* bp = kbh + (size_t)(k0 + kpair) * LL + (tileN + m16);
        v2f b;
        b.x = bp[0];
        b.y = bp[LL];
        acc = __builtin_amdgcn_wmma_f32_16x16x4_f32(
            /*neg_a=*/false, a, /*neg_b=*/false, b,
            /*c_mod=*/(short)0, acc, /*reuse_a=*/false, /*reuse_b=*/false);
    }

    const float scale = 0.125f;                    // 1/sqrt(64)
    const int   nCol  = tileN + (lane & 15);
    const int   mBase = tileM + ((lane >> 4) << 3);
#pragma unroll
    for (int r = 0; r < 8; ++r)
        sbh[(size_t)(mBase + r) * LL + nCol] = acc[r] * scale;
}

// ---------------------------------------------------------------------------
// Kernel 2: softmax across the HEAD axis (reference: nn.Softmax() on 4D ->
// dim=1). One thread per (b,i,j); strided by L*L across the 8 head planes.
// In-place on the score region.
// ---------------------------------------------------------------------------
__global__ void softmax_over_heads(float* __restrict__ score)
{
    const size_t plane = (size_t)LL * LL;
    size_t idx = (size_t)blockIdx.x * blockDim.x + threadIdx.x;  // over B*L*L
    if (idx >= (size_t)BB * plane) return;
    const size_t b  = idx / plane;
    const size_t ij = idx - b * plane;
    float* p = score + b * (size_t)HH * plane + ij;

    float vals[HH];
    float m = -1e30f;
#pragma unroll
    for (int h = 0; h < HH; ++h) { vals[h] = p[(size_t)h * plane]; m = fmaxf(m, vals[h]); }
    float s = 0.f;
#pragma unroll
    for (int h = 0; h < HH; ++h) { vals[h] = __expf(vals[h] - m); s += vals[h]; }
    const float inv = 1.f / s;
#pragma unroll
    for (int h = 0; h < HH; ++h) p[(size_t)h * plane] = vals[h] * inv;
}

// ---------------------------------------------------------------------------
// Kernel 3: stable top-NUM (ascending) per score row + gather of v rows.
// One 256-thread block per (b,h,i). Scores staged in LDS; rank-based stable
// selection reproduces jnp.argsort(...)[...,-NUM:] semantics exactly:
//   rank(j) = #{y < x} + #{jj < j : y == x}; keep ranks >= L-NUM.
// Gather writes float4 (v rows are 64-float aligned).
// ---------------------------------------------------------------------------
__global__ void __launch_bounds__(256)
topk_gather(const float* __restrict__ score, const float* __restrict__ v,
            float* __restrict__ out)
{
    __shared__ float sv[LL];
    __shared__ int   sidx[NUM];

    const int row = blockIdx.x;                    // 0..B*H*L-1
    const int bh  = row / LL;
    const int tid = threadIdx.x;
    const float* srow = score + (size_t)row * LL;

    for (int j = tid; j < LL; j += 256) sv[j] = srow[j];
    __syncthreads();

    for (int j = tid; j < LL; j += 256) {
        const float x = sv[j];
        int rank = 0;
        for (int jj = 0; jj < LL; ++jj) {
            const float y = sv[jj];
            rank += (y < x) || ((y == x) && (jj < j));
        }
        if (rank >= LL - NUM) sidx[rank - (LL - NUM)] = j;
    }
    __syncthreads();

    const float4* vbh  = (const float4*)(v + (size_t)bh * LL * DD);
    float4*       orow = (float4*)(out + (size_t)row * NUM * DD);
    for (int e = tid; e < NUM * (DD / 4); e += 256) {
        const int t = e >> 4;                      // which of the 100 indices
        const int c = e & 15;                      // which float4 of the row
        orow[e] = vbh[(size_t)sidx[t] * (DD / 4) + c];
    }
}

// ---------------------------------------------------------------------------
extern "C" void kernel_launch(void* const* d_in, const int* in_sizes, int n_in,
                              void* d_out, int out_size, void* d_ws, size_t ws_size,
                              hipStream_t stream)
{
    (void)in_sizes; (void)n_in; (void)d_ws; (void)ws_size; (void)out_size;
    const float* q = (const float*)d_in[0];
    const float* k = (const float*)d_in[1];
    const float* v = (const float*)d_in[2];
    // d_in[3] is `num` (==100); shapes are compile-time constants here.

    float* out   = (float*)d_out;                               // [B,H,L,NUM,D]
    float* score = out + (size_t)BB * HH * LL * NUM * DD;       // [B,H,L,L]

    // 1) fp32 WMMA GEMM: 48x48 tiles per (b,h), 4 waves/block.
    {
        constexpr int TILES = LL / 16;                          // 48
        dim3 grid((TILES * TILES) / 4, BB * HH);
        qk_gemm_wmma_f32<<<grid, 128, 0, stream>>>(q, k, score);
    }
    // 2) softmax over heads, in place.
    {
        const size_t n = (size_t)BB * LL * LL;
        softmax_over_heads<<<(unsigned)((n + 255) / 256), 256, 0, stream>>>(score);
    }
    // 3) top-100 + gather.
    {
        topk_gather<<<BB * HH * LL, 256, 0, stream>>>(score, v, out);
    }
}